// WindowAttentionWithViewAwareRoPE_34591666602379
// MI455X (gfx1250) — compile-verified
//
#include <hip/hip_runtime.h>
#include <hip/hip_bf16.h>
#include <math.h>

// ---------------------------------------------------------------------------
// WindowAttention with view-aware RoPE for MI455X (gfx1250, wave32, WMMA).
// Memory-bound problem (~155 GFLOP vs ~2 GB traffic) -> keep full fp32 and
// use V_WMMA_F32_16X16X4_F32 for every GEMM stage.
// ---------------------------------------------------------------------------

typedef float v2f __attribute__((ext_vector_type(2)));
typedef float v4f __attribute__((ext_vector_type(4)));
typedef float v8f __attribute__((ext_vector_type(8)));

#define DIM      256
#define NHEAD    8
#define HD       32
#define SEQ      64
#define NW       64
#define NBATCH   4096
#define MAX_LOGIT 4.6051701859880914f   // log(1/0.01)
#define LOG2_10000_DIV16 0.83048202372184058696f

#define XPITCH 260   // 64x256 x-tile pitch (bank-conflict-free A-frag column reads)
#define SPITCH 33    // per-wave 64x32 q/k staging pitch
#define VPITCH 36    // 64x32 q/k/v tile pitch in attention kernel (16B aligned rows)
#define PPITCH 68    // 16x64 probability strip pitch

// --------------------------- WMMA wrapper ----------------------------------
__device__ __forceinline__ v8f wmma_f32(v2f a, v2f b, v8f c) {
  // D = A(16x4) * B(4x16) + C(16x16), fp32
  return __builtin_amdgcn_wmma_f32_16x16x4_f32(false, a, false, b, (short)0, c,
                                               false, false);
}

// 64x16 output tile (4 M-tiles stacked) of  X(64x256) @ W^T, W row-major (o,c),
// X staged in LDS with pitch XPITCH.  K = 256 -> 64 k-blocks of 4.
__device__ __forceinline__ void gemm_tile_64x16(const float* __restrict__ sX,
                                                const float* __restrict__ w,
                                                int o0, int l16, int lhi,
                                                v8f acc[4]) {
#pragma unroll
  for (int m = 0; m < 4; ++m) acc[m] = {};
  const float* wrow = w + (size_t)(o0 + l16) * DIM + 2 * lhi;
  const float* a0   = sX + l16 * XPITCH + 2 * lhi;
#pragma unroll 4
  for (int kb = 0; kb < 64; ++kb) {
    v2f bfrag = *(const v2f*)(wrow + 4 * kb);
#pragma unroll
    for (int m = 0; m < 4; ++m) {
      v2f afrag = *(const v2f*)(a0 + m * 16 * XPITCH + 4 * kb);
      acc[m] = wmma_f32(afrag, bfrag, acc[m]);
    }
  }
}

// RoPE + L2-normalize one head's 64x32 strip (in per-wave LDS staging),
// write to dst_base[(n)*HD + d].  scale folds exp(min(logit_scale,MAX)).
__device__ __forceinline__ void rope_norm_store(const float* __restrict__ stg,
                                                float* __restrict__ dst_base,
                                                float scale, int lane) {
#pragma unroll
  for (int rr = 0; rr < 2; ++rr) {
    const int n = lane + rr * 32;
    const float* rowp = stg + n * SPITCH;
    const float t = (float)n + 0.1f;  // VIEW_ID * VIEW_OFFSET = 0.1
    float outv[32];
    float ss = 0.f;
#pragma unroll
    for (int i = 0; i < 16; ++i) {
      float inv_freq = exp2f(-(float)i * LOG2_10000_DIV16);
      float fr = t * inv_freq;
      float s, c;
      __sincosf(fr, &s, &c);
      float x1 = rowp[i], x2 = rowp[i + 16];
      float o1 = x1 * c - x2 * s;
      float o2 = x1 * s + x2 * c;
      outv[i] = o1;
      outv[i + 16] = o2;
      ss = fmaf(o1, o1, fmaf(o2, o2, ss));
    }
    float mul = scale * rsqrtf(fmaxf(ss, 1e-24f));  // == /max(||q||,1e-12)
    float* dst = dst_base + (size_t)n * HD;
#pragma unroll
    for (int i = 0; i < 32; i += 4) {
      v4f vv = {outv[i] * mul, outv[i + 1] * mul, outv[i + 2] * mul,
                outv[i + 3] * mul};
      *(v4f*)(dst + i) = vv;
    }
  }
}

// =========================== Kernel 1: QKV =================================
// grid = 4096 windows, block = 256 (8 waves).  Wave w owns head w for q and k
// and v columns [32w, 32w+32).  Dynamic LDS: x-tile + 8 per-wave stagings.
__global__ void __launch_bounds__(256) qkv_rope_norm_kernel(
    const float* __restrict__ x, const float* __restrict__ qkv_w,
    const float* __restrict__ qkv_b, const float* __restrict__ logit_scale,
    float* __restrict__ qn, float* __restrict__ kn, float* __restrict__ vv) {
  extern __shared__ float smem[];
  float* sX = smem;                        // 64 x XPITCH
  float* sStage = smem + SEQ * XPITCH;     // 8 x (64 x SPITCH)

  const int b = blockIdx.x;
  const int tid = threadIdx.x;
  const int wave = tid >> 5;
  const int lane = tid & 31;
  const int l16 = lane & 15;
  const int lhi = lane >> 4;

  // ---- stage x[b] (64x256) into LDS ----
  const float* xb = x + (size_t)b * SEQ * DIM;
  for (int i = tid; i < SEQ * DIM / 4; i += 256) {
    int row = i >> 6;
    int c4 = i & 63;
    *(v4f*)(sX + row * XPITCH + c4 * 4) = *(const v4f*)(xb + row * DIM + c4 * 4);
  }
  __syncthreads();

  float* stg = sStage + wave * SEQ * SPITCH;

  // ---- Q (head = wave) ----
  {
    v8f acc[4];
#pragma unroll
    for (int nt = 0; nt < 2; ++nt) {
      int o0 = wave * HD + nt * 16;  // q section starts at output col 0
      gemm_tile_64x16(sX, qkv_w, o0, l16, lhi, acc);
      float bias = qkv_b[o0 + l16];
#pragma unroll
      for (int m = 0; m < 4; ++m)
#pragma unroll
        for (int r = 0; r < 8; ++r)
          stg[(m * 16 + r + 8 * lhi) * SPITCH + nt * 16 + l16] = acc[m][r] + bias;
    }
  }
  __syncthreads();
  {
    float scale = __expf(fminf(logit_scale[wave], MAX_LOGIT));
    float* dst = qn + (((size_t)b * NHEAD + wave) * SEQ) * HD;
    rope_norm_store(stg, dst, scale, lane);
  }
  __syncthreads();

  // ---- K (head = wave) ----
  {
    v8f acc[4];
#pragma unroll
    for (int nt = 0; nt < 2; ++nt) {
      int o0 = DIM + wave * HD + nt * 16;  // k section at col 256
      gemm_tile_64x16(sX, qkv_w, o0, l16, lhi, acc);
      float bias = qkv_b[o0 + l16];
#pragma unroll
      for (int m = 0; m < 4; ++m)
#pragma unroll
        for (int r = 0; r < 8; ++r)
          stg[(m * 16 + r + 8 * lhi) * SPITCH + nt * 16 + l16] = acc[m][r] + bias;
    }
  }
  __syncthreads();
  {
    float* dst = kn + (((size_t)b * NHEAD + wave) * SEQ) * HD;
    rope_norm_store(stg, dst, 1.0f, lane);
  }

  // ---- V (cols 32w..32w+31 of v section; head = wave) ----
  {
    v8f acc[4];
    float* vb = vv + (((size_t)b * NHEAD + wave) * SEQ) * HD;
#pragma unroll
    for (int nt = 0; nt < 2; ++nt) {
      int o0 = 2 * DIM + wave * HD + nt * 16;  // v section at col 512
      gemm_tile_64x16(sX, qkv_w, o0, l16, lhi, acc);
      float bias = qkv_b[o0 + l16];
#pragma unroll
      for (int m = 0; m < 4; ++m)
#pragma unroll
        for (int r = 0; r < 8; ++r)
          vb[(size_t)(m * 16 + r + 8 * lhi) * HD + nt * 16 + l16] =
              acc[m][r] + bias;
    }
  }
}

// ========================= Kernel 2: attention =============================
// grid = 4096*8 (b,h) pairs, block = 128 (4 waves, one 16-row strip each).
__global__ void __launch_bounds__(128) attn_kernel(
    const float* __restrict__ qn, const float* __restrict__ kn,
    const float* __restrict__ vv, const float* __restrict__ mask,
    float* __restrict__ outH) {
  __shared__ float sQ[SEQ * VPITCH];
  __shared__ float sK[SEQ * VPITCH];
  __shared__ float sV[SEQ * VPITCH];
  __shared__ float sP[4 * 16 * PPITCH];

  const int bh = blockIdx.x;
  const int b = bh >> 3;
  const int h = bh & 7;
  const int wi = b & (NW - 1);  // window index = b % nW
  const int tid = threadIdx.x;
  const int wave = tid >> 5;
  const int lane = tid & 31;
  const int l16 = lane & 15;
  const int lhi = lane >> 4;

  // ---- stage qn/kn/v head tiles (64x32 each) ----
  const size_t base = ((size_t)b * NHEAD + h) * SEQ * HD;
  for (int i = tid; i < SEQ * HD / 4; i += 128) {
    int row = i >> 3;
    int c4 = i & 7;
    *(v4f*)(sQ + row * VPITCH + c4 * 4) = *(const v4f*)(qn + base + row * HD + c4 * 4);
    *(v4f*)(sK + row * VPITCH + c4 * 4) = *(const v4f*)(kn + base + row * HD + c4 * 4);
    *(v4f*)(sV + row * VPITCH + c4 * 4) = *(const v4f*)(vv + base + row * HD + c4 * 4);
  }
  __syncthreads();

  // ---- S = q @ k^T for 16-row strip (K = 32 -> 8 k-blocks) ----
  v8f acc[4] = {};
  const float* aq = sQ + (wave * 16 + l16) * VPITCH + 2 * lhi;
#pragma unroll
  for (int kb = 0; kb < 8; ++kb) {
    v2f afrag = *(const v2f*)(aq + 4 * kb);
#pragma unroll
    for (int nt = 0; nt < 4; ++nt) {
      v2f bfrag = *(const v2f*)(sK + (nt * 16 + l16) * VPITCH + 2 * lhi + 4 * kb);
      acc[nt] = wmma_f32(afrag, bfrag, acc[nt]);
    }
  }

  // ---- + mask, spill strip to LDS for softmax / re-layout ----
  const float* mrow = mask + (size_t)wi * SEQ * SEQ;
  float* pst = sP + wave * 16 * PPITCH;
#pragma unroll
  for (int nt = 0; nt < 4; ++nt)
#pragma unroll
    for (int r = 0; r < 8; ++r) {
      int rloc = r + 8 * lhi;
      int n = wave * 16 + rloc;
      int m = nt * 16 + l16;
      pst[rloc * PPITCH + m] = acc[nt][r] + mrow[n * SEQ + m];
    }
  __syncthreads();

  // ---- softmax: lane owns (row l16, half lhi); pair-combine via shfl_xor ----
  {
    float* prow = pst + l16 * PPITCH + lhi * 32;
    float vals[32];
    float pm = -1e30f;
#pragma unroll
    for (int i = 0; i < 32; ++i) {
      vals[i] = prow[i];
      pm = fmaxf(pm, vals[i]);
    }
    pm = fmaxf(pm, __shfl_xor(pm, 16, 32));
    float psum = 0.f;
#pragma unroll
    for (int i = 0; i < 32; ++i) {
      vals[i] = __expf(vals[i] - pm);
      psum += vals[i];
    }
    psum += __shfl_xor(psum, 16, 32);
    float rinv = 1.0f / psum;
#pragma unroll
    for (int i = 0; i < 32; ++i) prow[i] = vals[i] * rinv;
  }
  __syncthreads();

  // ---- O = P @ V  (16x64 @ 64x32, K = 64 -> 16 k-blocks) ----
  v8f oacc[2] = {};
  const float* ap = pst + l16 * PPITCH + 2 * lhi;
#pragma unroll
  for (int kb = 0; kb < 16; ++kb) {
    v2f afrag = *(const v2f*)(ap + 4 * kb);
    int k0 = 4 * kb + 2 * lhi;
#pragma unroll
    for (int nt = 0; nt < 2; ++nt) {
      v2f bfrag;
      bfrag.x = sV[(k0 + 0) * VPITCH + nt * 16 + l16];
      bfrag.y = sV[(k0 + 1) * VPITCH + nt * 16 + l16];
      oacc[nt] = wmma_f32(afrag, bfrag, oacc[nt]);
    }
  }

  // ---- write strip to outH[b][n][h*32+d] (head-transposed layout) ----
  float* ob = outH + (size_t)b * SEQ * DIM + h * HD;
#pragma unroll
  for (int nt = 0; nt < 2; ++nt)
#pragma unroll
    for (int r = 0; r < 8; ++r) {
      int n = wave * 16 + r + 8 * lhi;
      ob[(size_t)n * DIM + nt * 16 + l16] = oacc[nt][r];
    }
}

// ======================== Kernel 3: projection =============================
// grid = 4096, block = 256.  out = outH @ proj_w^T + proj_b.
__global__ void __launch_bounds__(256) proj_kernel(
    const float* __restrict__ outH, const float* __restrict__ proj_w,
    const float* __restrict__ proj_b, float* __restrict__ out) {
  extern __shared__ float smem[];
  float* sX = smem;  // 64 x XPITCH

  const int b = blockIdx.x;
  const int tid = threadIdx.x;
  const int wave = tid >> 5;
  const int lane = tid & 31;
  const int l16 = lane & 15;
  const int lhi = lane >> 4;

  const float* xb = outH + (size_t)b * SEQ * DIM;
  for (int i = tid; i < SEQ * DIM / 4; i += 256) {
    int row = i >> 6;
    int c4 = i & 63;
    *(v4f*)(sX + row * XPITCH + c4 * 4) = *(const v4f*)(xb + row * DIM + c4 * 4);
  }
  __syncthreads();

  float* ob = out + (size_t)b * SEQ * DIM;
#pragma unroll
  for (int nt = 0; nt < 2; ++nt) {
    int o0 = wave * 32 + nt * 16;
    v8f acc[4];
    gemm_tile_64x16(sX, proj_w, o0, l16, lhi, acc);
    float bias = proj_b[o0 + l16];
#pragma unroll
    for (int m = 0; m < 4; ++m)
#pragma unroll
      for (int r = 0; r < 8; ++r)
        ob[(size_t)(m * 16 + r + 8 * lhi) * DIM + o0 + l16] = acc[m][r] + bias;
  }
}

// ============================ host launcher ================================
extern "C" void kernel_launch(void* const* d_in, const int* in_sizes, int n_in,
                              void* d_out, int out_size, void* d_ws,
                              size_t ws_size, hipStream_t stream) {
  const float* x = (const float*)d_in[0];
  const float* mask = (const float*)d_in[1];
  const float* qkv_w = (const float*)d_in[2];
  const float* qkv_b = (const float*)d_in[3];
  const float* proj_w = (const float*)d_in[4];
  const float* proj_b = (const float*)d_in[5];
  const float* logit_scale = (const float*)d_in[6];
  float* out = (float*)d_out;

  float* ws = (float*)d_ws;
  const size_t per = (size_t)NBATCH * NHEAD * SEQ * HD;  // 64M floats
  float* qn = ws;
  float* kn = ws + per;
  float* vv = ws + 2 * per;
  float* oh = ws + 3 * per;

  const size_t lds1 = (size_t)(SEQ * XPITCH + 8 * SEQ * SPITCH) * sizeof(float);
  qkv_rope_norm_kernel<<<NBATCH, 256, lds1, stream>>>(x, qkv_w, qkv_b,
                                                      logit_scale, qn, kn, vv);

  attn_kernel<<<NBATCH * NHEAD, 128, 0, stream>>>(qn, kn, vv, mask, oh);

  const size_t lds3 = (size_t)(SEQ * XPITCH) * sizeof(float);
  proj_kernel<<<NBATCH, 256, lds3, stream>>>(oh, proj_w, proj_b, out);
}